// inv_loss_73358041415726
// MI455X (gfx1250) — compile-verified
//
#include <hip/hip_runtime.h>
#include <hip/hip_bf16.h>
#include <cstdint>
#include <math.h>

// ---------------------------------------------------------------------------
// 2D FFT magnitude-L1 loss for k[16,3,1024,1024] fp32 on gfx1250 (MI455X).
// Pass 1: row FFTs (radix-16 first stage via V_WMMA_F32_16X16X4_F32, then
//         3x Stockham radix-4 in LDS), store transposed, kx<=512 only
//         (conjugate symmetry of real-input row FFT).
// Pass 2: column FFTs (complex), sum of magnitudes, weight 2x for mirrored
//         columns, deterministic block reduction.
// Pass 3: deterministic final reduction -> out[0] = total / 48.
// Data staging uses CDNA5 async global->LDS loads + s_wait_asynccnt.
// ---------------------------------------------------------------------------

typedef __attribute__((ext_vector_type(2))) float v2f;
typedef __attribute__((ext_vector_type(8))) float v8f;

#define PI_F 3.14159265358979323846f

#define N1024       1024
#define NPLANES     48
#define ROWS_PER_WG 8
#define ROWBLOCKS   (N1024 / ROWS_PER_WG)   // 128
#define KCOLS       513                      // kx = 0..512 kept
#define NTHREADS    128

// ---------------- helpers ---------------------------------------------------

__device__ __forceinline__ float2 cmul(float2 a, float2 b) {
  return make_float2(a.x * b.x - a.y * b.y, a.x * b.y + a.y * b.x);
}

// CDNA5 async copy: 16 bytes global memory -> LDS, tracked by ASYNCcnt.
__device__ __forceinline__ void async_ld_b128(unsigned lds_byte_off, const void* g) {
  unsigned long long ga = (unsigned long long)(uintptr_t)g;
  asm volatile("global_load_async_to_lds_b128 %0, %1, off"
               :: "v"(lds_byte_off), "v"(ga) : "memory");
}
__device__ __forceinline__ void async_wait0() {
  asm volatile("s_wait_asynccnt 0" ::: "memory");
}

__device__ __forceinline__ v8f wmma4(v2f a, v2f b, v8f c) {
  // D(16x16) += A(16x4) * B(4x16), f32
  return __builtin_amdgcn_wmma_f32_16x16x4_f32(false, a, false, b, (short)0, c,
                                               false, false);
}

// DFT16 matrix A-fragments per WMMA A layout (f32 16x4):
//   lanes 0-15:  vgpr0 = A[m][K=0], vgpr1 = A[m][K=1]
//   lanes 16-31: vgpr0 = A[m][K=2], vgpr1 = A[m][K=3]
// D[m][n1] = exp(-2*pi*i*m*n1/16): dr = cos, di = sin (angle negative).
__device__ __forceinline__ void make_dft16_frags(int lane, v2f dr[4], v2f di[4]) {
  const int m    = lane & 15;
  const int koff = (lane >> 4) << 1;  // 0 or 2
#pragma unroll
  for (int kk = 0; kk < 4; ++kk) {
    const int n1a = 4 * kk + koff;
    const int n1b = n1a + 1;
    const float pa = -(PI_F / 8.0f) * (float)((m * n1a) & 15);
    const float pb = -(PI_F / 8.0f) * (float)((m * n1b) & 15);
    float sa, ca, sb, cb;
    __sincosf(pa, &sa, &ca);
    __sincosf(pb, &sb, &cb);
    dr[kk].x = ca; dr[kk].y = cb;
    di[kk].x = sa; di[kk].y = sb;
  }
}

// One Stockham radix-4 stage applied to 16 independent 64-point FFTs living in
// segments of 64 complex values. 256 butterflies total, 2 per thread.
template <int Ns>
__device__ __forceinline__ void stockham_stage(const float2* __restrict__ src,
                                               float2* __restrict__ dst, int tid) {
#pragma unroll
  for (int it = 0; it < 2; ++it) {
    const int bj  = tid + it * NTHREADS;  // 0..255
    const int seg = bj >> 4;              // which 64-pt FFT
    const int j   = bj & 15;              // butterfly id in segment
    const float2* s = src + (seg << 6);
    float2 a0 = s[j];
    float2 a1 = s[j + 16];
    float2 a2 = s[j + 32];
    float2 a3 = s[j + 48];
    const int jm = j & (Ns - 1);
    const float ang = -2.0f * PI_F * (float)jm / (float)(4 * Ns);
    float sw, cw;
    __sincosf(ang, &sw, &cw);
    const float2 w1 = make_float2(cw, sw);
    const float2 w2 = cmul(w1, w1);
    const float2 w3 = cmul(w2, w1);
    a1 = cmul(a1, w1);
    a2 = cmul(a2, w2);
    a3 = cmul(a3, w3);
    const float2 t0 = make_float2(a0.x + a2.x, a0.y + a2.y);
    const float2 t1 = make_float2(a0.x - a2.x, a0.y - a2.y);
    const float2 t2 = make_float2(a1.x + a3.x, a1.y + a3.y);
    const float2 t3 = make_float2(a1.x - a3.x, a1.y - a3.y);
    const float2 b0 = make_float2(t0.x + t2.x, t0.y + t2.y);
    const float2 b2 = make_float2(t0.x - t2.x, t0.y - t2.y);
    const float2 b1 = make_float2(t1.x + t3.y, t1.y - t3.x);  // t1 - i*t3
    const float2 b3 = make_float2(t1.x - t3.y, t1.y + t3.x);  // t1 + i*t3
    const int idxD = (j / Ns) * (4 * Ns) + jm;
    float2* d = dst + (seg << 6);
    d[idxD]          = b0;
    d[idxD + Ns]     = b1;
    d[idxD + 2 * Ns] = b2;
    d[idxD + 3 * Ns] = b3;
  }
}

// ---------------- pass 1: row FFTs ------------------------------------------

__global__ __launch_bounds__(NTHREADS) void fft_rows_kernel(
    const float* __restrict__ gin, float2* __restrict__ T) {
  __shared__ __align__(16) float  s_in[N1024];
  __shared__ __align__(16) float2 s_a[N1024];
  __shared__ __align__(16) float2 s_b[N1024];
  __shared__ __align__(16) float2 s_stag[KCOLS * ROWS_PER_WG];

  const int tid  = threadIdx.x;
  const int lane = tid & 31;
  const int wv   = tid >> 5;
  const int m    = lane & 15;
  const int sel  = lane >> 4;
  const int koff = sel << 1;
  const int n2   = (wv << 4) + m;   // this wave's n2-tile column

  const int p  = blockIdx.y;        // plane 0..47
  const int rb = blockIdx.x;        // row block 0..127
  const int y0 = rb * ROWS_PER_WG;

  v2f dr[4], di[4];
  make_dft16_frags(lane, dr, di);

  const unsigned lds_in = (unsigned)(uintptr_t)(&s_in[0]);
  const float* grow0 = gin + ((size_t)p * N1024 + y0) * N1024;

  // async prefetch of row 0 (2 x b128 per lane = 4 KB)
  {
    const char* g = (const char*)grow0;
    async_ld_b128(lds_in + tid * 32,      g + tid * 32);
    async_ld_b128(lds_in + tid * 32 + 16, g + tid * 32 + 16);
  }

  for (int r = 0; r < ROWS_PER_WG; ++r) {
    async_wait0();
    __syncthreads();  // row r resident in s_in for all waves

    // --- radix-16 first stage via WMMA: Y(16x64) = DFT16 * Xmat -------------
    // Xmat[n1][n2] = x[64*n1 + n2]; wave wv computes n2 in [16*wv, 16*wv+16).
    v8f cr = {0.f, 0.f, 0.f, 0.f, 0.f, 0.f, 0.f, 0.f};
    v8f ci = cr;
#pragma unroll
    for (int kk = 0; kk < 4; ++kk) {
      v2f b;
      b.x = s_in[((4 * kk + koff)     << 6) + n2];
      b.y = s_in[((4 * kk + koff + 1) << 6) + n2];
      cr = wmma4(dr[kk], b, cr);   // Yr = Dr * X
      ci = wmma4(di[kk], b, ci);   // Yi = Di * X   (real input)
    }
    __syncthreads();  // all waves done reading s_in

    if (r + 1 < ROWS_PER_WG) {  // overlap next row's DMA with compute
      const char* g = (const char*)(grow0 + (size_t)(r + 1) * N1024);
      async_ld_b128(lds_in + tid * 32,      g + tid * 32);
      async_ld_b128(lds_in + tid * 32 + 16, g + tid * 32 + 16);
    }

    // --- twiddle W_1024^{k1*n2}, scatter to s_a[k1*64 + n2] -----------------
#pragma unroll
    for (int v = 0; v < 8; ++v) {
      const int k1 = v + (sel << 3);  // C-layout: M = v (+8 for lanes 16-31)
      const float ang = -2.0f * PI_F * (float)(k1 * n2) * (1.0f / 1024.0f);
      float sw, cw;
      __sincosf(ang, &sw, &cw);
      s_a[(k1 << 6) + n2] = cmul(make_float2(cr[v], ci[v]), make_float2(cw, sw));
    }
    __syncthreads();

    // --- 16 independent 64-pt FFTs over n2 (Stockham radix-4, autosort) -----
    stockham_stage<1>(s_a, s_b, tid);   __syncthreads();
    stockham_stage<4>(s_b, s_a, tid);   __syncthreads();
    stockham_stage<16>(s_a, s_b, tid);  __syncthreads();

    // slot s = k1*64 + k2 holds frequency k = k1 + 16*k2; keep k <= 512
#pragma unroll
    for (int i = 0; i < 8; ++i) {
      const int s  = tid + (i << 7);
      const int k1 = s >> 6;
      const int k2 = s & 63;
      const int k  = k1 + (k2 << 4);
      if (k < KCOLS) s_stag[k * ROWS_PER_WG + r] = s_b[s];
    }
    __syncthreads();
  }

  // transposed, coalesced write-out: T[p][k][y0..y0+7]
  for (int e = tid; e < KCOLS * ROWS_PER_WG; e += NTHREADS) {
    const int k  = e >> 3;
    const int rr = e & 7;
    T[((size_t)p * KCOLS + k) * N1024 + (y0 + rr)] = s_stag[e];
  }
}

// ---------------- pass 2: column FFTs + magnitude sum -----------------------

__global__ __launch_bounds__(NTHREADS) void fft_cols_kernel(
    const float2* __restrict__ T, float* __restrict__ partials) {
  __shared__ __align__(16) float2 s_in[N1024];
  __shared__ __align__(16) float2 s_a[N1024];
  __shared__ __align__(16) float2 s_b[N1024];
  __shared__ float s_red[NTHREADS];

  const int tid  = threadIdx.x;
  const int lane = tid & 31;
  const int wv   = tid >> 5;
  const int m    = lane & 15;
  const int sel  = lane >> 4;
  const int koff = sel << 1;
  const int n2   = (wv << 4) + m;

  const int kx = blockIdx.x;   // 0..512
  const int p  = blockIdx.y;   // 0..47

  v2f dr[4], di[4], ndi[4];
  make_dft16_frags(lane, dr, di);
#pragma unroll
  for (int kk = 0; kk < 4; ++kk) { ndi[kk].x = -di[kk].x; ndi[kk].y = -di[kk].y; }

  // async load of the full complex column (8 KB, 4 x b128 per lane)
  const unsigned lds_in = (unsigned)(uintptr_t)(&s_in[0]);
  const char* g = (const char*)(T + ((size_t)p * KCOLS + kx) * N1024);
#pragma unroll
  for (int i = 0; i < 4; ++i)
    async_ld_b128(lds_in + tid * 64 + i * 16, g + tid * 64 + i * 16);
  async_wait0();
  __syncthreads();

  // --- complex radix-16 first stage via WMMA --------------------------------
  v8f cr = {0.f, 0.f, 0.f, 0.f, 0.f, 0.f, 0.f, 0.f};
  v8f ci = cr;
#pragma unroll
  for (int kk = 0; kk < 4; ++kk) {
    const float2 xa = s_in[((4 * kk + koff)     << 6) + n2];
    const float2 xb = s_in[((4 * kk + koff + 1) << 6) + n2];
    v2f br, bi;
    br.x = xa.x; br.y = xb.x;
    bi.x = xa.y; bi.y = xb.y;
    cr = wmma4(dr[kk],  br, cr);   // Yr += Dr*Xr
    cr = wmma4(ndi[kk], bi, cr);   // Yr -= Di*Xi
    ci = wmma4(di[kk],  br, ci);   // Yi += Di*Xr
    ci = wmma4(dr[kk],  bi, ci);   // Yi += Dr*Xi
  }
  __syncthreads();

#pragma unroll
  for (int v = 0; v < 8; ++v) {
    const int k1 = v + (sel << 3);
    const float ang = -2.0f * PI_F * (float)(k1 * n2) * (1.0f / 1024.0f);
    float sw, cw;
    __sincosf(ang, &sw, &cw);
    s_a[(k1 << 6) + n2] = cmul(make_float2(cr[v], ci[v]), make_float2(cw, sw));
  }
  __syncthreads();

  stockham_stage<1>(s_a, s_b, tid);   __syncthreads();
  stockham_stage<4>(s_b, s_a, tid);   __syncthreads();
  stockham_stage<16>(s_a, s_b, tid);  __syncthreads();

  // --- sum of magnitudes (output ordering irrelevant) -----------------------
  float loc = 0.f;
#pragma unroll
  for (int i = 0; i < 8; ++i) {
    const float2 z = s_b[tid + (i << 7)];
    loc += sqrtf(z.x * z.x + z.y * z.y);
  }
  s_red[tid] = loc;
  __syncthreads();
#pragma unroll
  for (int off = NTHREADS / 2; off > 0; off >>= 1) {
    if (tid < off) s_red[tid] += s_red[tid + off];
    __syncthreads();
  }
  if (tid == 0) {
    const float wgt = (kx == 0 || kx == 512) ? 1.0f : 2.0f;  // conj symmetry
    partials[(size_t)p * KCOLS + kx] = s_red[0] * wgt;
  }
}

// ---------------- pass 3: deterministic final reduction ---------------------

__global__ __launch_bounds__(256) void final_reduce_kernel(
    const float* __restrict__ partials, float* __restrict__ out) {
  __shared__ float sp[256];
  const int tid = threadIdx.x;
  const int n = NPLANES * KCOLS;
  float acc = 0.f;
  for (int i = tid; i < n; i += 256) acc += partials[i];
  sp[tid] = acc;
  __syncthreads();
#pragma unroll
  for (int off = 128; off > 0; off >>= 1) {
    if (tid < off) sp[tid] += sp[tid + off];
    __syncthreads();
  }
  if (tid == 0) out[0] = sp[0] * (1.0f / 48.0f);  // mean over 16*3 plane-sums
}

// ---------------- host launcher ---------------------------------------------

extern "C" void kernel_launch(void* const* d_in, const int* in_sizes, int n_in,
                              void* d_out, int out_size, void* d_ws, size_t ws_size,
                              hipStream_t stream) {
  (void)in_sizes; (void)n_in; (void)out_size; (void)ws_size;
  const float* kin = (const float*)d_in[0];               // [48][1024][1024] f32
  float2* T = (float2*)d_ws;                              // [48][513][1024] c64
  const size_t t_bytes = (size_t)NPLANES * KCOLS * N1024 * sizeof(float2);
  float* partials = (float*)((char*)d_ws + t_bytes);      // [48*513] f32

  fft_rows_kernel<<<dim3(ROWBLOCKS, NPLANES), NTHREADS, 0, stream>>>(kin, T);
  fft_cols_kernel<<<dim3(KCOLS, NPLANES), NTHREADS, 0, stream>>>(T, partials);
  final_reduce_kernel<<<1, 256, 0, stream>>>(partials, (float*)d_out);
}